// NSHE_65223373357672
// MI455X (gfx1250) — compile-verified
//
#include <hip/hip_runtime.h>
#include <math.h>

typedef __attribute__((ext_vector_type(2))) float v2f;
typedef __attribute__((ext_vector_type(8))) float v8f;

#define NA_N 4353
#define NM_N 3676
#define NT_N 8000
#define NN   (NA_N + NM_N + NT_N)   // 16029
#define E_N  500000
#define S_N  200000

// ---------------------------------------------------------------------------
// Generic f32 WMMA GEMM:  C[M x Nc] = act(A[M x K] * B[K x Nc] + bias)
// One wave computes one 16x16 output tile via V_WMMA_F32_16X16X4_F32.
// Requires: K % 4 == 0, Nc % 16 == 0. M handled with clamped loads + guarded
// stores (EXEC stays all-ones through every WMMA).
// ---------------------------------------------------------------------------
__global__ __launch_bounds__(256) void gemm_wmma_f32(
    const float* __restrict__ A, const float* __restrict__ B,
    const float* __restrict__ bias, float* __restrict__ C,
    int M, int K, int Nc, int relu) {
  int wid  = (int)((blockIdx.x * blockDim.x + threadIdx.x) >> 5);
  int lane = threadIdx.x & 31;
  int tilesN = Nc >> 4;
  int tilesM = (M + 15) >> 4;
  if (wid >= tilesM * tilesN) return;          // wave-uniform
  int tm = wid / tilesN, tn = wid % tilesN;

  int r      = lane & 15;           // row (A) or col (B/store) within tile
  int khalf  = (lane >> 4) << 1;    // lanes 16-31 handle K+2,K+3
  int arow   = tm * 16 + r; if (arow >= M) arow = M - 1;   // clamp, keep EXEC full
  const float* aptr = A + (size_t)arow * K + khalf;
  const float* bptr = B + (size_t)khalf * Nc + (tn * 16 + r);

  v8f c = {};
#pragma unroll 4
  for (int k0 = 0; k0 < K; k0 += 4) {
    v2f a = { aptr[0], aptr[1] };          // contiguous 8B
    v2f b = { bptr[0], bptr[Nc] };         // stride-Nc pair
    c = __builtin_amdgcn_wmma_f32_16x16x4_f32(false, a, false, b,
                                              (short)0, c, false, false);
    aptr += 4;
    bptr += (size_t)4 * Nc;
  }

  int   is_hi = lane >> 4;
  float bv = bias ? bias[tn * 16 + r] : 0.0f;
#pragma unroll
  for (int rr = 0; rr < 8; rr++) {
    int row = tm * 16 + rr + is_hi * 8;
    float v = c[rr] + bv;
    if (relu) v = fmaxf(v, 0.0f);
    if (row < M) C[(size_t)row * Nc + tn * 16 + r] = v;
  }
}

// ---------------------------------------------------------------------------
// segment_sum scatter: out[dst[e]][:] += X[src[e]][:]   (D = 256 or 128)
// One thread handles one float4 of one edge -> coalesced global_atomic_add_f32
// ---------------------------------------------------------------------------
__global__ __launch_bounds__(256) void scatter_add(
    const float* __restrict__ X, const int* __restrict__ src,
    const int* __restrict__ dst, float* __restrict__ out, int E, int D) {
  int tpe = D >> 2;                               // float4s per edge
  long long gid = (long long)blockIdx.x * blockDim.x + threadIdx.x;
  long long e = gid / tpe;
  int t = (int)(gid - e * tpe);
  if (e >= E) return;
  int s = src[e], d = dst[e];
  float4 v = ((const float4*)(X + (size_t)s * D))[t];
  float* o = out + (size_t)d * D + (size_t)t * 4;
  atomicAdd(o + 0, v.x);
  atomicAdd(o + 1, v.y);
  atomicAdd(o + 2, v.z);
  atomicAdd(o + 3, v.w);
}

// ---------------------------------------------------------------------------
// Row L2 normalize (D=128), in place; also emits the movie slice (output 3).
// One wave per row, 4 floats/lane, wave32 shfl_xor reduction.
// ---------------------------------------------------------------------------
__global__ __launch_bounds__(256) void normalize_rows(
    float* __restrict__ h, float* __restrict__ out_movie, int M) {
  int row  = (int)((blockIdx.x * blockDim.x + threadIdx.x) >> 5);
  int lane = threadIdx.x & 31;
  if (row >= M) return;
  float4 v = ((const float4*)(h + (size_t)row * 128))[lane];
  float s = v.x * v.x + v.y * v.y + v.z * v.z + v.w * v.w;
#pragma unroll
  for (int off = 16; off >= 1; off >>= 1) s += __shfl_xor(s, off, 32);
  float scale = 1.0f / fmaxf(sqrtf(s), 1e-12f);
  v.x *= scale; v.y *= scale; v.z *= scale; v.w *= scale;
  ((float4*)(h + (size_t)row * 128))[lane] = v;
  if (row >= NA_N && row < NA_N + NM_N)
    ((float4*)(out_movie + (size_t)(row - NA_N) * 128))[lane] = v;
}

// ---------------------------------------------------------------------------
// Classifier: for 16 samples per wave, gather concat(seg0[i0][0:128],
// seg1[i1][0:64], seg2[i2][0:64]) as the A fragment stream of a 16x16x256
// WMMA GEMM against Wh (256x16); then hid = relu(.), p = hid . Wo,
// x = sigmoid(p). Hidden reduction across each 16-lane half via shfl_xor.
// ---------------------------------------------------------------------------
__global__ __launch_bounds__(256) void cls_wmma(
    const float* __restrict__ seg0, const float* __restrict__ seg1,
    const float* __restrict__ seg2,
    const int* __restrict__ i0, const int* __restrict__ i1,
    const int* __restrict__ i2,
    const float* __restrict__ Wh, const float* __restrict__ Wo,
    float* __restrict__ xout, int S) {
  int wid  = (int)((blockIdx.x * blockDim.x + threadIdx.x) >> 5);
  int lane = threadIdx.x & 31;
  int base = wid * 16;
  if (base >= S) return;                        // wave-uniform

  int r = lane & 15;
  int m = base + r; if (m >= S) m = S - 1;      // clamp keeps EXEC full
  const float* p0 = seg0 + (size_t)i0[m] * 128;
  const float* p1 = seg1 + (size_t)i1[m] * 64;
  const float* p2 = seg2 + (size_t)i2[m] * 64;
  int khalf = (lane >> 4) << 1;

  v8f c = {};
#pragma unroll
  for (int k0 = 0; k0 < 256; k0 += 4) {         // each 4-step fits one segment
    const float* ap; int col;
    if (k0 < 128)      { ap = p0; col = k0; }
    else if (k0 < 192) { ap = p1; col = k0 - 128; }
    else               { ap = p2; col = k0 - 192; }
    v2f a = { ap[col + khalf], ap[col + khalf + 1] };
    int kb = k0 + khalf;
    v2f b = { Wh[kb * 16 + r], Wh[(kb + 1) * 16 + r] };
    c = __builtin_amdgcn_wmma_f32_16x16x4_f32(false, a, false, b,
                                              (short)0, c, false, false);
  }

  float wo  = Wo[r];                            // n = lane&15 hidden index
  float myp = 0.0f;
#pragma unroll
  for (int rr = 0; rr < 8; rr++) {
    float v = fmaxf(c[rr], 0.0f) * wo;          // relu(hid) * Wo[n]
    v += __shfl_xor(v, 1, 32);
    v += __shfl_xor(v, 2, 32);
    v += __shfl_xor(v, 4, 32);
    v += __shfl_xor(v, 8, 32);                  // sum over n within 16-lane half
    if (r == rr) myp = v;                       // lane rr: row rr (+8 in hi half)
  }
  if (r < 8) {
    int mm = base + r + (lane >> 4) * 8;
    if (mm < S) xout[mm] = 1.0f / (1.0f + expf(-myp));
  }
}

// ---------------------------------------------------------------------------
static inline void launch_gemm(const float* A, const float* B, const float* bias,
                               float* C, int M, int K, int Nc, int relu,
                               hipStream_t stream) {
  int tiles = ((M + 15) >> 4) * (Nc >> 4);
  int blocks = (tiles + 7) / 8;                 // 8 waves / 256-thread block
  gemm_wmma_f32<<<blocks, 256, 0, stream>>>(A, B, bias, C, M, K, Nc, relu);
}

extern "C" void kernel_launch(void* const* d_in, const int* in_sizes, int n_in,
                              void* d_out, int out_size, void* d_ws, size_t ws_size,
                              hipStream_t stream) {
  (void)in_sizes; (void)n_in; (void)out_size; (void)ws_size;
  const float* h_author  = (const float*)d_in[0];
  const float* h_movie   = (const float*)d_in[1];
  const float* h_term    = (const float*)d_in[2];
  const float* Wp_author = (const float*)d_in[3];
  const float* Wp_movie  = (const float*)d_in[4];
  const float* Wp_term   = (const float*)d_in[5];
  const float* W1        = (const float*)d_in[6];
  const float* b1        = (const float*)d_in[7];
  const float* W2        = (const float*)d_in[8];
  const float* b2        = (const float*)d_in[9];
  const float* Wc_author = (const float*)d_in[10];
  const float* Wc_movie  = (const float*)d_in[11];
  const float* Wc_term   = (const float*)d_in[12];
  const float* Wh_author = (const float*)d_in[13];
  const float* Wh_movie  = (const float*)d_in[14];
  const float* Wh_term   = (const float*)d_in[15];
  const float* Wo_author = (const float*)d_in[16];
  const float* Wo_movie  = (const float*)d_in[17];
  const float* Wo_term   = (const float*)d_in[18];
  const int*   src       = (const int*)d_in[19];
  const int*   dst       = (const int*)d_in[20];
  const int*   ns0_a     = (const int*)d_in[21];
  const int*   ns0_m     = (const int*)d_in[22];
  const int*   ns0_t     = (const int*)d_in[23];
  const int*   ns1_a     = (const int*)d_in[24];
  const int*   ns1_m     = (const int*)d_in[25];
  const int*   ns1_t     = (const int*)d_in[26];
  const int*   ns2_a     = (const int*)d_in[27];
  const int*   ns2_m     = (const int*)d_in[28];
  const int*   ns2_t     = (const int*)d_in[29];

  // d_out layout: h (NN*128) | x (3*S) | h_movie slice (NM*128)
  float* h_out  = (float*)d_out;
  float* x_out  = h_out + (size_t)NN * 128;
  float* hm_out = x_out + (size_t)3 * S_N;

  // Workspace: bufA = H -> h1 -> {ca,cm,ct};  bufB = agg1 -> agg2
  float* bufA = (float*)d_ws;                    // NN*256 floats
  float* bufB = bufA + (size_t)NN * 256;         // NN*256 floats
  float* H    = bufA;
  float* agg1 = bufB;
  float* h1   = bufA;
  float* agg2 = bufB;
  float* ca   = bufA;
  float* cm   = ca + (size_t)NA_N * 64;
  float* ct   = cm + (size_t)NM_N * 64;

  // 1) Projections -> H (NN x 256)
  launch_gemm(h_author, Wp_author, nullptr, H,                         NA_N, 512, 256, 0, stream);
  launch_gemm(h_movie,  Wp_movie,  nullptr, H + (size_t)NA_N * 256,    NM_N, 512, 256, 0, stream);
  launch_gemm(h_term,   Wp_term,   nullptr, H + (size_t)(NA_N+NM_N)*256, NT_N, 512, 256, 0, stream);

  // 2) agg1 = segment_sum(H[src], dst)
  hipMemsetAsync(agg1, 0, (size_t)NN * 256 * sizeof(float), stream);
  {
    long long thr = (long long)E_N * 64;
    scatter_add<<<(int)((thr + 255) / 256), 256, 0, stream>>>(H, src, dst, agg1, E_N, 256);
  }

  // 3) h1 = relu(agg1 @ W1 + b1)   (NN x 128)
  launch_gemm(agg1, W1, b1, h1, NN, 256, 128, 1, stream);

  // 4) agg2 = segment_sum(h1[src], dst)
  hipMemsetAsync(agg2, 0, (size_t)NN * 128 * sizeof(float), stream);
  {
    long long thr = (long long)E_N * 32;
    scatter_add<<<(int)((thr + 255) / 256), 256, 0, stream>>>(h1, src, dst, agg2, E_N, 128);
  }

  // 5) h2 = agg2 @ W2 + b2 -> d_out h region, then L2-normalize in place
  launch_gemm(agg2, W2, b2, h_out, NN, 128, 128, 0, stream);
  {
    int blocks = (NN + 7) / 8;                   // 8 rows (waves) per block
    normalize_rows<<<blocks, 256, 0, stream>>>(h_out, hm_out, NN);
  }

  // 6) c-embeddings: ca/cm/ct = h_slice @ Wc_*  (128 -> 64)
  const float* ha = h_out;
  const float* hm = h_out + (size_t)NA_N * 128;
  const float* ht = h_out + (size_t)(NA_N + NM_N) * 128;
  launch_gemm(ha, Wc_author, nullptr, ca, NA_N, 128, 64, 0, stream);
  launch_gemm(hm, Wc_movie,  nullptr, cm, NM_N, 128, 64, 0, stream);
  launch_gemm(ht, Wc_term,   nullptr, ct, NT_N, 128, 64, 0, stream);

  // 7) Classifiers (gathered-A WMMA) + sigmoid
  int cls_blocks = ((S_N + 15) / 16 + 7) / 8;    // waves of 16 samples, 8 waves/block
  cls_wmma<<<cls_blocks, 256, 0, stream>>>(ha, cm, ct, ns0_a, ns0_m, ns0_t,
                                           Wh_author, Wo_author, x_out,          S_N);
  cls_wmma<<<cls_blocks, 256, 0, stream>>>(hm, ca, ct, ns1_m, ns1_a, ns1_t,
                                           Wh_movie,  Wo_movie,  x_out + S_N,    S_N);
  cls_wmma<<<cls_blocks, 256, 0, stream>>>(ht, ca, cm, ns2_t, ns2_a, ns2_m,
                                           Wh_term,   Wo_term,   x_out + 2*S_N,  S_N);
}